// GATMLP_SCST_wgt_3521873182758
// MI455X (gfx1250) — compile-verified
//
#include <hip/hip_runtime.h>
#include <math.h>

typedef __attribute__((ext_vector_type(2))) float v2f;
typedef __attribute__((ext_vector_type(8))) float v8f;

#define NHEADS 3
#define CDIM   64
#define HC     192   // NHEADS*CDIM
#define FDIM   256
#define HIDDIM 128
#define NEG_SLOPE 0.2f

// ---------------- helpers ----------------
__device__ inline float atomicMaxFloat(float* addr, float value) {
  // valid given addr initialized to -inf; sign-split trick
  if (value >= 0.0f)
    return __int_as_float(atomicMax((int*)addr, __float_as_int(value)));
  return __uint_as_float(atomicMin((unsigned int*)addr, __float_as_uint(value)));
}

// ---------------- per-tag init ----------------
__global__ void k_init_tag(float* __restrict__ og, float* __restrict__ mbuf,
                           float* __restrict__ denom, float* __restrict__ z, int N) {
  int i = blockIdx.x * blockDim.x + threadIdx.x;
  if (i < N * HC) og[i] = 0.0f;
  if (i < N * NHEADS) { mbuf[i] = -INFINITY; denom[i] = 0.0f; }
  if (i < FDIM) z[i] = 0.0f;
}

// ---------------- WMMA f32 GEMM: C[N x OC] = A[N x K] * B[K x OC] ----------------
// Block = 4 waves; each wave computes a full 16 x OC row-strip (NCT = OC/16 tiles),
// reusing one A fragment across all column tiles per K-step.
// B strips of 32 K-rows are staged into LDS with gfx1250 async global->LDS copies
// (ASYNCcnt) and consumed via ds_load. mode 1: C = gelu(acc + bias[col]).
template<int K, int OC>
__global__ void k_wmma_gemm_f32(const float* __restrict__ A, const float* __restrict__ B,
                                const float* __restrict__ bias, float* __restrict__ C,
                                int Nrows, int mode)
{
  constexpr int NCT = OC / 16;           // column tiles per wave
  __shared__ float bsh[32 * OC];         // B strip: 32 K-rows x OC cols

  int tid  = threadIdx.y * 32 + threadIdx.x;   // 0..127
  int lane = threadIdx.x;                      // 0..31 (wave32)
  int half = lane >> 4;                        // 0: lanes 0-15, 1: lanes 16-31
  int l15  = lane & 15;
  int koff = half << 1;                        // K sub-offset 0 or 2 (ISA 16x4 A layout)

  int rowTile = blockIdx.x * 4 + (int)threadIdx.y;
  int maxTile = (Nrows >> 4) - 1;
  if (rowTile > maxTile) rowTile = maxTile;    // clamp tail: keep all waves in barriers
  int row0 = rowTile * 16;
  const float* arow = A + (size_t)(row0 + l15) * K;

  v8f acc[NCT];
#pragma unroll
  for (int ct = 0; ct < NCT; ++ct) acc[ct] = v8f{};

  unsigned ldsBase = (unsigned)(uintptr_t)(&bsh[0]);   // LDS byte offset of staging buffer

  for (int kc = 0; kc < K; kc += 32) {
    // ---- async stage B[kc..kc+32) x OC into LDS (128 threads x NCT b128 copies) ----
#pragma unroll
    for (int i = 0; i < NCT; ++i) {
      int elem = (i * 128 + tid) * 4;                       // float index in strip
      unsigned ldsOff = ldsBase + (unsigned)(elem << 2);    // LDS byte address
      unsigned long long gaddr =
          (unsigned long long)(uintptr_t)(B + (size_t)kc * OC + elem);
      asm volatile("global_load_async_to_lds_b128 %0, %1, off"
                   :: "v"(ldsOff), "v"(gaddr) : "memory");
    }
    asm volatile("s_wait_asynccnt 0x0" ::: "memory");
    __syncthreads();

    // ---- 8 K-steps of 4, one A fragment feeds NCT WMMAs ----
#pragma unroll
    for (int k = 0; k < 32; k += 4) {
      v2f a;
      a.x = arow[kc + k + koff];
      a.y = arow[kc + k + koff + 1];
#pragma unroll
      for (int ct = 0; ct < NCT; ++ct) {
        v2f b;
        b.x = bsh[(k + koff) * OC + ct * 16 + l15];
        b.y = bsh[(k + koff + 1) * OC + ct * 16 + l15];
        acc[ct] = __builtin_amdgcn_wmma_f32_16x16x4_f32(false, a, false, b, (short)0,
                                                        acc[ct], false, false);
      }
    }
    __syncthreads();
  }

  // C/D layout: VGPR j -> M = j (lanes 0-15) or j+8 (lanes 16-31), N = lane&15
  int mbase = half << 3;
#pragma unroll
  for (int ct = 0; ct < NCT; ++ct) {
    int col = ct * 16 + l15;
#pragma unroll
    for (int j = 0; j < 8; ++j) {
      float v = acc[ct][j];
      if (mode == 1) {
        v += bias[col];
        v = 0.5f * v * (1.0f + erff(v * 0.70710678118654752f));  // exact gelu
      }
      C[(size_t)(row0 + mbase + j) * OC + col] = v;
    }
  }
}

// ---------------- attention scores: one wave per node ----------------
__global__ void k_scores(const float* __restrict__ h, const float* __restrict__ a_src,
                         const float* __restrict__ a_dst, float* __restrict__ s_src,
                         float* __restrict__ s_dst, int N)
{
  int wid  = blockIdx.x * (blockDim.x >> 5) + (threadIdx.x >> 5);
  int lane = threadIdx.x & 31;
  if (wid >= N) return;
  const float* hrow = h + (size_t)wid * HC;
#pragma unroll
  for (int head = 0; head < NHEADS; ++head) {
    float vs = 0.0f, vd = 0.0f;
    for (int c = lane; c < CDIM; c += 32) {
      float hv = hrow[head * CDIM + c];
      vs += hv * a_src[head * CDIM + c];
      vd += hv * a_dst[head * CDIM + c];
    }
    for (int off = 16; off; off >>= 1) {
      vs += __shfl_xor(vs, off, 32);
      vd += __shfl_xor(vd, off, 32);
    }
    if (lane == 0) { s_src[wid * NHEADS + head] = vs; s_dst[wid * NHEADS + head] = vd; }
  }
}

__device__ inline void edge_endpoints(const int* __restrict__ ei, int E, int i,
                                      int& src, int& dst) {
  if (i < E) { src = ei[i]; dst = ei[E + i]; }
  else       { src = i - E; dst = i - E; }       // self loop
}

// ---------------- edge pass 1: segment max ----------------
__global__ void k_edge_max(const int* __restrict__ ei, int E, int N,
                           const float* __restrict__ s_src, const float* __restrict__ s_dst,
                           float* __restrict__ mbuf)
{
  int i = blockIdx.x * blockDim.x + threadIdx.x;
  if (i >= E + N) return;
  int src, dst; edge_endpoints(ei, E, i, src, dst);
#pragma unroll
  for (int hd = 0; hd < NHEADS; ++hd) {
    float e = s_src[src * NHEADS + hd] + s_dst[dst * NHEADS + hd];
    e = (e > 0.0f) ? e : NEG_SLOPE * e;
    atomicMaxFloat(&mbuf[dst * NHEADS + hd], e);
  }
}

// ---------------- edge pass 2: segment sum of exp(e - m) ----------------
__global__ void k_edge_sum(const int* __restrict__ ei, int E, int N,
                           const float* __restrict__ s_src, const float* __restrict__ s_dst,
                           const float* __restrict__ mbuf, float* __restrict__ denom)
{
  int i = blockIdx.x * blockDim.x + threadIdx.x;
  if (i >= E + N) return;
  int src, dst; edge_endpoints(ei, E, i, src, dst);
#pragma unroll
  for (int hd = 0; hd < NHEADS; ++hd) {
    float e = s_src[src * NHEADS + hd] + s_dst[dst * NHEADS + hd];
    e = (e > 0.0f) ? e : NEG_SLOPE * e;
    atomicAdd(&denom[dst * NHEADS + hd], expf(e - mbuf[dst * NHEADS + hd]));
  }
}

// ---------------- edge pass 3: weighted aggregation (one wave per edge) ----------------
__global__ void k_edge_aggr(const int* __restrict__ ei, int E, int N,
                            const float* __restrict__ s_src, const float* __restrict__ s_dst,
                            const float* __restrict__ mbuf, const float* __restrict__ denom,
                            const float* __restrict__ h, float* __restrict__ og)
{
  int wid  = (blockIdx.x * blockDim.x + threadIdx.x) >> 5;
  int lane = threadIdx.x & 31;
  if (wid >= E + N) return;
  int src, dst; edge_endpoints(ei, E, wid, src, dst);
  float alpha[NHEADS];
#pragma unroll
  for (int hd = 0; hd < NHEADS; ++hd) {
    float e = s_src[src * NHEADS + hd] + s_dst[dst * NHEADS + hd];
    e = (e > 0.0f) ? e : NEG_SLOPE * e;
    alpha[hd] = expf(e - mbuf[dst * NHEADS + hd]) / (denom[dst * NHEADS + hd] + 1e-16f);
  }
  const float* hs = h + (size_t)src * HC;
  float* od = og + (size_t)dst * HC;
  for (int j = lane; j < HC; j += 32)
    atomicAdd(&od[j], alpha[j >> 6] * hs[j]);   // L2-resident atomics
}

// ---------------- elu(x + bias), in place ----------------
__global__ void k_elu_bias(float* __restrict__ og, const float* __restrict__ bias, int N)
{
  int i = blockIdx.x * blockDim.x + threadIdx.x;
  if (i >= N * HC) return;
  float v = og[i] + bias[i % HC];
  og[i] = (v > 0.0f) ? v : expm1f(v);
}

// ---------------- beta = (hid . w2 + b2)/100, one wave per node ----------------
__global__ void k_beta(const float* __restrict__ hid, const float* __restrict__ w2,
                       const float* __restrict__ b2, float* __restrict__ beta, int N)
{
  int wid  = (blockIdx.x * blockDim.x + threadIdx.x) >> 5;
  int lane = threadIdx.x & 31;
  if (wid >= N) return;
  const float* hr = hid + (size_t)wid * HIDDIM;
  float acc = 0.0f;
  for (int c = lane; c < HIDDIM; c += 32) acc += hr[c] * w2[c];
  for (int off = 16; off; off >>= 1) acc += __shfl_xor(acc, off, 32);
  if (lane == 0) beta[wid] = (acc + b2[0]) * 0.01f;
}

// ---------------- z[f] = sum_n x[n][f] * beta[n] (coalesced, thread = feature) ----------------
__global__ void k_xtb(const float* __restrict__ x, const float* __restrict__ beta,
                      float* __restrict__ z, int N)
{
  int f = threadIdx.x;                                 // 256 features
  int chunk = (N + gridDim.x - 1) / gridDim.x;
  int n0 = blockIdx.x * chunk;
  int n1 = n0 + chunk; if (n1 > N) n1 = N;
  float acc = 0.0f;
  for (int n = n0; n < n1; ++n)
    acc += x[(size_t)n * FDIM + f] * beta[n];
  atomicAdd(&z[f], acc);
}

// ---------------- out[f] = sigmoid(z_sc + 0.5*z_st) ----------------
__global__ void k_final(const float* __restrict__ z_sc, const float* __restrict__ z_st,
                        float* __restrict__ out)
{
  int f = threadIdx.x;
  float t = z_sc[f] + 0.5f * z_st[f];
  out[f] = 1.0f / (1.0f + expf(-t));
}

// ---------------- host orchestration ----------------
static void run_tag(const float* x, const int* ei,
                    const float* wg, const float* a_src, const float* a_dst, const float* bg,
                    const float* w1, const float* b1, const float* w2, const float* b2,
                    float* hbuf, float* og, float* ssrc, float* sdst, float* mb, float* den,
                    float* hid, float* z, float* beta_out, int N, int E, hipStream_t stream)
{
  int totNC = N * HC;
  k_init_tag<<<(totNC + 255) / 256, 256, 0, stream>>>(og, mb, den, z, N);

  dim3 gb(32, 4);
  int rowTiles = (N + 15) / 16;
  int gx = (rowTiles + 3) / 4;
  k_wmma_gemm_f32<FDIM, HC><<<gx, gb, 0, stream>>>(x, wg, nullptr, hbuf, N, 0);

  k_scores<<<(N + 7) / 8, 256, 0, stream>>>(hbuf, a_src, a_dst, ssrc, sdst, N);

  int Etot = E + N;
  k_edge_max <<<(Etot + 255) / 256, 256, 0, stream>>>(ei, E, N, ssrc, sdst, mb);
  k_edge_sum <<<(Etot + 255) / 256, 256, 0, stream>>>(ei, E, N, ssrc, sdst, mb, den);
  k_edge_aggr<<<(Etot + 7) / 8,    256, 0, stream>>>(ei, E, N, ssrc, sdst, mb, den, hbuf, og);

  k_elu_bias<<<(totNC + 255) / 256, 256, 0, stream>>>(og, bg, N);

  k_wmma_gemm_f32<HC, HIDDIM><<<gx, gb, 0, stream>>>(og, w1, b1, hid, N, 1);

  k_beta<<<(N + 7) / 8, 256, 0, stream>>>(hid, w2, b2, beta_out, N);
  k_xtb<<<256, 256, 0, stream>>>(x, beta_out, z, N);
}

extern "C" void kernel_launch(void* const* d_in, const int* in_sizes, int n_in,
                              void* d_out, int out_size, void* d_ws, size_t ws_size,
                              hipStream_t stream)
{
  const float* x_st  = (const float*)d_in[0];
  const int*   ei_st = (const int*)  d_in[1];
  const float* x_sc  = (const float*)d_in[2];
  const int*   ei_sc = (const int*)  d_in[3];
  int N = in_sizes[0] / FDIM;
  int E = in_sizes[1] / 2;

  // st params: 4..11, sc params: 12..19
  const float* p_st[8]; const float* p_sc[8];
  for (int i = 0; i < 8; ++i) { p_st[i] = (const float*)d_in[4 + i];
                                p_sc[i] = (const float*)d_in[12 + i]; }

  // workspace carve (256B aligned)
  char* wp = (char*)d_ws;
  auto take = [&](size_t bytes) -> float* {
    float* r = (float*)wp;
    wp += (bytes + 255) & ~(size_t)255;
    return r;
  };
  float* hbuf = take((size_t)N * HC * 4);
  float* og   = take((size_t)N * HC * 4);
  float* ssrc = take((size_t)N * NHEADS * 4);
  float* sdst = take((size_t)N * NHEADS * 4);
  float* mb   = take((size_t)N * NHEADS * 4);
  float* den  = take((size_t)N * NHEADS * 4);
  float* hid  = take((size_t)N * HIDDIM * 4);
  float* z_st = take(FDIM * 4);
  float* z_sc = take(FDIM * 4);

  float* out_main = (float*)d_out;        // [256]
  float* beta_st  = out_main + FDIM;      // [N]
  float* beta_sc  = beta_st + N;          // [N]

  run_tag(x_st, ei_st, p_st[0], p_st[1], p_st[2], p_st[3], p_st[4], p_st[5], p_st[6], p_st[7],
          hbuf, og, ssrc, sdst, mb, den, hid, z_st, beta_st, N, E, stream);
  run_tag(x_sc, ei_sc, p_sc[0], p_sc[1], p_sc[2], p_sc[3], p_sc[4], p_sc[5], p_sc[6], p_sc[7],
          hbuf, og, ssrc, sdst, mb, den, hid, z_sc, beta_sc, N, E, stream);

  k_final<<<1, 256, 0, stream>>>(z_sc, z_st, out_main);
}